// Model_39676907882989
// MI455X (gfx1250) — compile-verified
//
#include <hip/hip_runtime.h>
#include <hip/hip_bf16.h>

typedef __attribute__((ext_vector_type(2))) float v2f;
typedef __attribute__((ext_vector_type(8))) float v8f;

#define DMODEL 32
#define KEEP_THRESH 13421773u  // keep if top-24-bit hash < floor(0.8 * 2^24)

// ---------------------------------------------------------------------------
// Kernel 1: fused linear projections.
//   q[r] = (x1[r] @ wq_w.T + wq_b) * (1/16)   [fold the score scale into q]
//   k[r] =  x2[r] @ wk_w.T + wk_b
//   v[r] =  x2[r] @ wv_w.T + wv_b
// ---------------------------------------------------------------------------
__global__ __launch_bounds__(256) void proj_kernel(
    const float* __restrict__ x1, const float* __restrict__ x2,
    const float* __restrict__ wq_w, const float* __restrict__ wq_b,
    const float* __restrict__ wk_w, const float* __restrict__ wk_b,
    const float* __restrict__ wv_w, const float* __restrict__ wv_b,
    float* __restrict__ q, float* __restrict__ k, float* __restrict__ v,
    int Nq, int Nk)
{
    int r = blockIdx.x * blockDim.x + threadIdx.x;
    if (r < Nq) {
        float x[DMODEL];
        #pragma unroll
        for (int d = 0; d < DMODEL; ++d) x[d] = x1[r * DMODEL + d];
        #pragma unroll
        for (int j = 0; j < 4; ++j) {
            float acc = wq_b[j];
            #pragma unroll
            for (int d = 0; d < DMODEL; ++d) acc = fmaf(x[d], wq_w[j * DMODEL + d], acc);
            q[r * 4 + j] = acc * 0.0625f;   // fold 1/INV_SCALE
        }
    }
    if (r < Nk) {
        float x[DMODEL];
        #pragma unroll
        for (int d = 0; d < DMODEL; ++d) x[d] = x2[r * DMODEL + d];
        #pragma unroll
        for (int j = 0; j < 4; ++j) {
            float acc = wk_b[j];
            #pragma unroll
            for (int d = 0; d < DMODEL; ++d) acc = fmaf(x[d], wk_w[j * DMODEL + d], acc);
            k[r * 4 + j] = acc;
        }
        #pragma unroll
        for (int j = 0; j < 8; ++j) {
            float acc = wv_b[j];
            #pragma unroll
            for (int d = 0; d < DMODEL; ++d) acc = fmaf(x[d], wv_w[j * DMODEL + d], acc);
            v[r * 8 + j] = acc;
        }
    }
}

__device__ __forceinline__ unsigned hash_u32(unsigned u) {
    u ^= u >> 16; u *= 0x7feb352du;
    u ^= u >> 15; u *= 0x846ca68bu;
    u ^= u >> 16;
    return u;
}

// ---------------------------------------------------------------------------
// Kernel 2: fused attention. One wave32 owns a 16-row Q tile.
// Pass 1: row max only (1 WMMA + 8 v_max per tile, no transcendentals).
// Pass 2: e = exp(s - m) once per element; Z accumulated pre-dropout; masked
//         weights transposed through per-wave LDS into A-layout; P@V via
//         4 WMMAs per 16-column K tile. Z sum-reduced after the loop.
// ---------------------------------------------------------------------------
__global__ __launch_bounds__(256) void attn_kernel(
    const float* __restrict__ q, const float* __restrict__ k,
    const float* __restrict__ v, float* __restrict__ out,
    int Nq, int Nk)
{
    __shared__ float lds[8 * 16 * 18];   // 8 waves x (16 rows x stride 18)

    const int lane = threadIdx.x & 31;
    const int wave = threadIdx.x >> 5;
    const int h    = lane >> 4;          // lane half (0/1)
    const int lm   = lane & 15;
    const int qtile = blockIdx.x * 8 + wave;
    const int qbase = qtile * 16;

    // A-layout Q tile (16x4 f32): lane L -> row lm, K-cols {2h, 2h+1}
    v2f a = *(const v2f*)(q + (size_t)(qbase + lm) * 4 + 2 * h);

    const int ntiles = Nk >> 4;

    // ---------------- Pass 1: per-row max (no exp needed) -------------------
    float rm[8];
    #pragma unroll
    for (int i = 0; i < 8; ++i) rm[i] = -1.0e30f;

    for (int t = 0; t < ntiles; ++t) {
        const int kbase = t * 16;
        // B-layout K^T tile (4x16): lane L -> col lm, rows {2h, 2h+1}
        v2f b = *(const v2f*)(k + (size_t)(kbase + lm) * 4 + 2 * h);
        __builtin_prefetch(k + (size_t)(kbase + 64 + lm) * 4, 0, 1);  // 4 tiles ahead
        v8f c = {};
        c = __builtin_amdgcn_wmma_f32_16x16x4_f32(false, a, false, b,
                                                  (short)0, c, false, false);
        #pragma unroll
        for (int i = 0; i < 8; ++i) rm[i] = fmaxf(rm[i], c[i]);
    }
    // max across the 16 lanes sharing each row (xor masks stay in-half)
    #pragma unroll
    for (int off = 1; off < 16; off <<= 1) {
        #pragma unroll
        for (int i = 0; i < 8; ++i) rm[i] = fmaxf(rm[i], __shfl_xor(rm[i], off));
    }

    // ---------------- Pass 2: Z + O in one sweep -----------------------------
    v8f o = {};
    float zs[8];
    #pragma unroll
    for (int i = 0; i < 8; ++i) zs[i] = 0.0f;
    float* myLds = lds + wave * (16 * 18);

    for (int t = 0; t < ntiles; ++t) {
        const int kbase = t * 16;
        v2f b = *(const v2f*)(k + (size_t)(kbase + lm) * 4 + 2 * h);
        __builtin_prefetch(k + (size_t)(kbase + 64 + lm) * 4, 0, 1);
        v8f c = {};
        c = __builtin_amdgcn_wmma_f32_16x16x4_f32(false, a, false, b,
                                                  (short)0, c, false, false);
        // exp once; Z accumulates PRE-dropout (softmax denominator);
        // masked weight goes to LDS in C-layout [M][N]
        #pragma unroll
        for (int i = 0; i < 8; ++i) {
            int row = qbase + i + 8 * h;
            int col = kbase + lm;
            float e = __expf(c[i] - rm[i]);
            zs[i] += e;
            unsigned hsh = hash_u32((unsigned)row * (unsigned)Nk + (unsigned)col);
            float w = ((hsh >> 8) < KEEP_THRESH) ? e : 0.0f;
            myLds[(i + 8 * h) * 18 + lm] = w;
        }
        // same-wave LDS ops are processed in order (ISA 7.3): only block
        // compiler reordering of the ds_load before the ds_store.
        asm volatile("" ::: "memory");

        // 4 K-chunks of 4 columns: A = P chunk (16x4), B = V rows (4x16, j<8)
        #pragma unroll
        for (int c4 = 0; c4 < 4; ++c4) {
            v2f aw = *(const v2f*)(myLds + lm * 18 + c4 * 4 + 2 * h);
            int r0 = kbase + c4 * 4 + 2 * h;
            v2f bv;
            bv.x = (lm < 8) ? v[(size_t)r0 * 8 + lm]       : 0.0f;
            bv.y = (lm < 8) ? v[(size_t)(r0 + 1) * 8 + lm] : 0.0f;
            o = __builtin_amdgcn_wmma_f32_16x16x4_f32(false, aw, false, bv,
                                                      (short)0, o, false, false);
        }
        asm volatile("" ::: "memory");   // WAR: loads before next tile's stores
    }

    // sum Z across the 16 lanes sharing each row
    #pragma unroll
    for (int off = 1; off < 16; off <<= 1) {
        #pragma unroll
        for (int i = 0; i < 8; ++i) zs[i] += __shfl_xor(zs[i], off);
    }

    // ---------------- Epilogue: scale by 1/(0.8 * Z_row) and store ----------
    if (lm < 8) {
        #pragma unroll
        for (int i = 0; i < 8; ++i) {
            int row = qbase + i + 8 * h;               // matches rm/zs/o row mapping
            out[(size_t)row * 8 + lm] = o[i] * (1.0f / (0.8f * zs[i]));
        }
    }
}

// ---------------------------------------------------------------------------
extern "C" void kernel_launch(void* const* d_in, const int* in_sizes, int n_in,
                              void* d_out, int out_size, void* d_ws, size_t ws_size,
                              hipStream_t stream)
{
    const float* x1   = (const float*)d_in[0];
    const float* x2   = (const float*)d_in[1];
    const float* wq_w = (const float*)d_in[2];
    const float* wq_b = (const float*)d_in[3];
    const float* wk_w = (const float*)d_in[4];
    const float* wk_b = (const float*)d_in[5];
    const float* wv_w = (const float*)d_in[6];
    const float* wv_b = (const float*)d_in[7];

    const int Nq = in_sizes[0] / DMODEL;   // 8192
    const int Nk = in_sizes[1] / DMODEL;   // 8192

    float* q = (float*)d_ws;               // Nq*4 f32
    float* k = q + (size_t)Nq * 4;         // Nk*4 f32
    float* v = k + (size_t)Nk * 4;         // Nk*8 f32

    const int nproj = (Nq > Nk) ? Nq : Nk;
    proj_kernel<<<(nproj + 255) / 256, 256, 0, stream>>>(
        x1, x2, wq_w, wq_b, wk_w, wk_b, wv_w, wv_b, q, k, v, Nq, Nk);

    // one wave per 16-row Q tile, 8 waves per block
    attn_kernel<<<Nq / 128, 256, 0, stream>>>(q, k, v, (float*)d_out, Nq, Nk);
}